// OxTM_82033875353949
// MI455X (gfx1250) — compile-verified
//
#include <hip/hip_runtime.h>
#include <hip/hip_bf16.h>
#include <math.h>

typedef __attribute__((ext_vector_type(2))) float v2f;
typedef __attribute__((ext_vector_type(8))) float v8f;

// ---------------- problem dims ----------------
constexpr int Bc = 4096;   // batch
constexpr int Vc = 2000;   // vocab (V1 == V2)
constexpr int Kc = 50;     // topics
constexpr int Hc = 500;    // hidden
constexpr int Ec = 300;    // embedding
constexpr int NP = 64;     // padded topic dim for WMMA tiles
constexpr int SINK_BLOCKS = 64;
constexpr int SINK_ITERS  = 500;

// ---------------- workspace layout (float offsets) ----------------
constexpr size_t OF_H1  = 0;
constexpr size_t OF_H2  = OF_H1 + (size_t)Bc * Hc;           // 2,048,000
constexpr size_t OF_MU  = OF_H2 + (size_t)Bc * Hc;
constexpr size_t OF_LV  = OF_MU + (size_t)Bc * Kc;
constexpr size_t OF_TH  = OF_LV + (size_t)Bc * Kc;
constexpr size_t OF_Y   = OF_TH + (size_t)Bc * Kc;           // 8,192,000 floats
constexpr size_t OF_MY  = OF_Y + (size_t)Bc * Vc;
constexpr size_t OF_VY  = OF_MY + 2048;
constexpr size_t OF_MMU = OF_VY + 2048;
constexpr size_t OF_VMU = OF_MMU + 64;
constexpr size_t OF_MLV = OF_VMU + 64;
constexpr size_t OF_VLV = OF_MLV + 64;
constexpr size_t OF_KLD = OF_VLV + 64;
constexpr size_t OF_LA  = OF_KLD + Bc;
constexpr size_t OF_LB  = OF_LA + Bc;
constexpr size_t OF_PART= OF_LB + Bc;
constexpr size_t OF_SYNC= OF_PART + 256;                     // ints live here
// time-sliced reuse (phase 2, after encoders/decoders are done):
constexpr size_t OF_M   = OF_Y;                              // 4,000,000
constexpr size_t OF_KM  = OF_Y + 4096000;                    // 4,000,000 (fits in y)
constexpr size_t OF_NA  = OF_H1;                             // 600,000
constexpr size_t OF_NB  = OF_H1 + (size_t)Vc * Ec;
constexpr size_t OF_AD  = OF_H2;                             // 4 x 128,000 in h2
constexpr size_t OF_BD  = OF_AD + (size_t)Vc * NP;
constexpr size_t OF_UD  = OF_BD + (size_t)Vc * NP;
constexpr size_t OF_VD  = OF_UD + (size_t)Vc * NP;

// ---------------- helpers ----------------
__device__ __forceinline__ v8f wmma4(v2f a, v2f b, v8f c) {
  return __builtin_amdgcn_wmma_f32_16x16x4_f32(false, a, false, b, (short)0, c, false, false);
}

__device__ __forceinline__ v2f loadv2(const float* p) {
  return *(const v2f*)p;   // 8B-aligned by construction (even K dims, even ka)
}

__device__ __forceinline__ float softplusf(float x) {
  return (x > 0.f) ? x + log1pf(expf(-x)) : log1pf(expf(x));
}

__device__ __forceinline__ float blockReduceSum256(float v, float* red) {
  int t = threadIdx.x;
  red[t] = v; __syncthreads();
  for (int s = 128; s > 0; s >>= 1) {
    if (t < s) red[t] += red[t + s];
    __syncthreads();
  }
  float r = red[0]; __syncthreads();
  return r;
}

__device__ __forceinline__ float blockReduceMax256(float v, float* red) {
  int t = threadIdx.x;
  red[t] = v; __syncthreads();
  for (int s = 128; s > 0; s >>= 1) {
    if (t < s) red[t] = fmaxf(red[t], red[t + s]);
    __syncthreads();
  }
  float r = red[0]; __syncthreads();
  return r;
}

__device__ __forceinline__ void grid_sync(int* cnt, int* gen, int nb) {
  __syncthreads();
  if (threadIdx.x == 0) {
    __threadfence();
    int g = __hip_atomic_load(gen, __ATOMIC_ACQUIRE, __HIP_MEMORY_SCOPE_AGENT);
    int prev = __hip_atomic_fetch_add(cnt, 1, __ATOMIC_ACQ_REL, __HIP_MEMORY_SCOPE_AGENT);
    if (prev == nb - 1) {
      __hip_atomic_store(cnt, 0, __ATOMIC_RELEASE, __HIP_MEMORY_SCOPE_AGENT);
      __hip_atomic_fetch_add(gen, 1, __ATOMIC_ACQ_REL, __HIP_MEMORY_SCOPE_AGENT);
    } else {
      while (__hip_atomic_load(gen, __ATOMIC_ACQUIRE, __HIP_MEMORY_SCOPE_AGENT) == g) {
        __builtin_amdgcn_s_sleep(2);
      }
    }
    __threadfence();
  }
  __syncthreads();
}

// ---------------- generic WMMA f32 GEMM ----------------
// mode: 0 plain store, 1 +bias, 2 +bias then softplus,
//       3 cosine: B is transposed (nb row-major, use nb^T), write M=1-c and Km=exp(-10*M)
// All callers have Mr % 16 == 0 and Kd even. Out-of-range output columns are
// computed against a clamped (valid) B column and simply never stored, which
// keeps the inner loop free of per-load exec masking.
__global__ void gemm_wmma(const float* __restrict__ A, const float* __restrict__ Bm,
                          const float* __restrict__ bias, float* __restrict__ C,
                          float* __restrict__ C2, int Mr, int Kd, int Nc, int mode) {
  int tid = blockIdx.x * 8 + (threadIdx.x >> 5);
  int lane = threadIdx.x & 31;
  int nt = (Nc + 15) >> 4;
  int strips = (Mr + 15) >> 4;
  if (tid >= strips * nt) return;              // wave-uniform: EXEC stays all-ones
  int i0 = (tid / nt) * 16;
  int n0 = (tid % nt) * 16;
  int m  = lane & 15;
  int hh = lane >> 4;
  int nn = n0 + m;
  int nc = (nn < Nc) ? nn : (Nc - 1);          // clamped column (stores still guarded)

  const float* Arow = A + (size_t)(i0 + m) * Kd + 2 * hh;
  int Kd4 = Kd & ~3;
  v8f c = {0.f,0.f,0.f,0.f,0.f,0.f,0.f,0.f};

  if (mode == 3) {                             // B^T: row nc of Bm, contiguous in k
    const float* Brow = Bm + (size_t)nc * Kd + 2 * hh;
    for (int j = 0; j < Kd4; j += 4) {
      v2f a = loadv2(Arow + j);
      v2f b = loadv2(Brow + j);
      c = wmma4(a, b, c);
    }
  } else {
    const float* Bp = Bm + (size_t)(2 * hh) * Nc + nc;
    for (int j = 0; j < Kd4; j += 4) {
      v2f a = loadv2(Arow + j);
      v2f b;
      b.x = Bp[(size_t)j * Nc];
      b.y = Bp[(size_t)(j + 1) * Nc];
      c = wmma4(a, b, c);
    }
  }
  if (Kd4 < Kd) {                              // K tail (only theta@phi, Kd=50)
    int ka = Kd4 + 2 * hh;
    v2f a, b;
    a.x = (ka     < Kd) ? Arow[Kd4]     : 0.f;
    a.y = (ka + 1 < Kd) ? Arow[Kd4 + 1] : 0.f;
    if (mode == 3) {
      b.x = (ka     < Kd) ? Bm[(size_t)nc * Kd + ka]     : 0.f;
      b.y = (ka + 1 < Kd) ? Bm[(size_t)nc * Kd + ka + 1] : 0.f;
    } else {
      b.x = (ka     < Kd) ? Bm[(size_t)ka * Nc + nc]       : 0.f;
      b.y = (ka + 1 < Kd) ? Bm[(size_t)(ka + 1) * Nc + nc] : 0.f;
    }
    c = wmma4(a, b, c);
  }

  for (int e = 0; e < 8; ++e) {
    int row = i0 + e + hh * 8;
    int col = n0 + m;
    if (col < Nc) {
      float val = c[e];
      if (mode == 1 || mode == 2) val += bias[col];
      if (mode == 2) val = softplusf(val);
      if (mode == 3) {
        float Mv = 1.f - val;
        C [(size_t)row * Nc + col] = Mv;
        C2[(size_t)row * Nc + col] = expf(-10.f * Mv);   // exp(-M/0.1)
      } else {
        C[(size_t)row * Nc + col] = val;
      }
    }
  }
}

// ---------------- column mean/var over batch (biased) ----------------
__global__ void k_colstats(const float* __restrict__ X, int R, int C,
                           float* __restrict__ mean, float* __restrict__ var) {
  int c = blockIdx.x * blockDim.x + threadIdx.x;
  if (c >= C) return;
  float s = 0.f, sq = 0.f;
  for (int r = 0; r < R; ++r) {
    float x = X[(size_t)r * C + c];
    s += x; sq += x * x;
  }
  float m = s / (float)R;
  mean[c] = m;
  var[c]  = sq / (float)R - m * m;
}

// ---------------- bn(mu/lv) -> z -> theta softmax + row KLD ----------------
__global__ void k_theta(const float* __restrict__ mu_raw, const float* __restrict__ lv_raw,
                        const float* __restrict__ mmu, const float* __restrict__ vmu,
                        const float* __restrict__ mlv, const float* __restrict__ vlv,
                        const float* __restrict__ mbn, const float* __restrict__ lbn,
                        const float* __restrict__ eps, float* __restrict__ theta,
                        float* __restrict__ rowkld) {
  int r = blockIdx.x, t = threadIdx.x;     // blockDim = 64
  __shared__ float sh[64];
  __shared__ float bc[2];
  bool act = (t < Kc);
  size_t o = (size_t)r * Kc + t;
  float mu = 0.f, lv = 0.f, z = 0.f;
  if (act) {
    mu = (mu_raw[o] - mmu[t]) * rsqrtf(vmu[t] + 1e-5f) + mbn[t];
    lv = (lv_raw[o] - mlv[t]) * rsqrtf(vlv[t] + 1e-5f) + lbn[t];
    z  = mu + expf(0.5f * lv) * eps[o];
  }
  sh[t] = act ? z : -3.4e38f;
  __syncthreads();
  if (t == 0) { float mx = sh[0]; for (int i = 1; i < Kc; ++i) mx = fmaxf(mx, sh[i]); bc[0] = mx; }
  __syncthreads();
  float mx = bc[0];
  sh[t] = act ? expf(z - mx) : 0.f;
  __syncthreads();
  if (t == 0) { float s = 0.f; for (int i = 0; i < Kc; ++i) s += sh[i]; bc[1] = s; }
  __syncthreads();
  float Z = bc[1];
  if (act) theta[o] = expf(z - mx) / Z;
  const float inv_var2 = 1.f / 0.98f;          // var2 = 1 - 2/K + 1/K
  const float log_var2 = -0.020202707f;        // ln(0.98)
  sh[t] = act ? (expf(lv) * inv_var2 + mu * mu * inv_var2 + log_var2 - lv) : 0.f;
  __syncthreads();
  if (t == 0) { float s = 0.f; for (int i = 0; i < Kc; ++i) s += sh[i]; rowkld[r] = 0.5f * (s - (float)Kc); }
}

// ---------------- bn(y) -> row softmax -> reconstruction loss ----------------
__global__ void k_rec(const float* __restrict__ y, const float* __restrict__ x,
                      const float* __restrict__ my, const float* __restrict__ vy,
                      const float* __restrict__ dbias, const float* __restrict__ rowkld,
                      float* __restrict__ rowloss) {
  __shared__ float red[256];
  int r = blockIdx.x, t = threadIdx.x;       // blockDim = 256, 8 cols/thread
  float s[8];
  float mx = -3.4e38f;
  for (int i = 0; i < 8; ++i) {
    int c = t + i * 256;
    float val = -3.4e38f;
    if (c < Vc) val = (y[(size_t)r * Vc + c] - my[c]) * rsqrtf(vy[c] + 1e-5f) + dbias[c];
    s[i] = val;
    mx = fmaxf(mx, val);
  }
  mx = blockReduceMax256(mx, red);
  float se = 0.f;
  for (int i = 0; i < 8; ++i) { int c = t + i * 256; if (c < Vc) se += expf(s[i] - mx); }
  float Z = blockReduceSum256(se, red);
  float acc = 0.f;
  for (int i = 0; i < 8; ++i) {
    int c = t + i * 256;
    if (c < Vc) acc += x[(size_t)r * Vc + c] * logf(expf(s[i] - mx) / Z + 1e-10f);
  }
  float rec = -blockReduceSum256(acc, red);
  if (t == 0) rowloss[r] = rec + rowkld[r];
}

// ---------------- row L2 normalize (BWE) ----------------
__global__ void k_rownorm(const float* __restrict__ X, float* __restrict__ out) {
  __shared__ float red[128];
  int r = blockIdx.x, t = threadIdx.x;       // blockDim = 128
  float sq = 0.f;
  for (int c = t; c < Ec; c += 128) { float v = X[(size_t)r * Ec + c]; sq += v * v; }
  red[t] = sq; __syncthreads();
  for (int s = 64; s > 0; s >>= 1) { if (t < s) red[t] += red[t + s]; __syncthreads(); }
  float d = fmaxf(sqrtf(red[0]), 1e-12f);
  __syncthreads();
  for (int c = t; c < Ec; c += 128) out[(size_t)r * Ec + c] = X[(size_t)r * Ec + c] / d;
}

// ---------------- softmax over vocab per topic; write padded (V x 64) ----------------
__global__ void k_colsoftmax(const float* __restrict__ phi, float* __restrict__ out) {
  __shared__ float red[256];
  int k = blockIdx.x, t = threadIdx.x;       // blockDim = 256
  const float* row = phi + (size_t)k * Vc;
  float mx = -3.4e38f;
  for (int v = t; v < Vc; v += 256) mx = fmaxf(mx, row[v]);
  mx = blockReduceMax256(mx, red);
  float se = 0.f;
  for (int v = t; v < Vc; v += 256) se += expf(row[v] - mx);
  float Z = blockReduceSum256(se, red);
  for (int v = t; v < Vc; v += 256) out[(size_t)v * NP + k] = expf(row[v] - mx) / Z;
}

// ---------------- init sinkhorn buffers (aD,bD,uD,vD zero; v = 1/V; sync/partials) ----
__global__ void k_initsink(float* __restrict__ base, float* __restrict__ part,
                           int* __restrict__ sync) {
  int idx = blockIdx.x * 256 + threadIdx.x;
  if (idx < 4 * Vc * NP) {
    float val = 0.f;
    if (idx >= 3 * Vc * NP) {                 // vD region
      int n = idx & (NP - 1);
      if (n < Kc) val = 1.f / (float)Vc;
    }
    base[idx] = val;
  }
  if (idx < 64) part[idx] = 0.f;
  if (idx < 4)  sync[idx] = 0;
}

// ---------------- persistent Sinkhorn kernel (500 iters, grid barrier) ----------------
__global__ void __launch_bounds__(256) k_sinkhorn(
    const float* __restrict__ Km, const float* __restrict__ Mm,
    const float* __restrict__ aD, const float* __restrict__ bD,
    float* __restrict__ uD, float* __restrict__ vD,
    float* __restrict__ part, int* __restrict__ sync) {
  __shared__ float Vsh[32 * NP];
  __shared__ float red[256];
  int wave = threadIdx.x >> 5, lane = threadIdx.x & 31;
  int task = blockIdx.x * 8 + wave;
  bool valid = task < 500;                   // wave-uniform
  int i0 = (task >> 2) * 16;
  int n0 = (task & 3) * 16;
  int m  = lane & 15;
  int hh = lane >> 4;
  int* cnt = sync;
  int* gen = sync + 1;
  const float* Arow  = Km + (size_t)(i0 + m) * Vc + 2 * hh;   // row-major row, phase A
  const float* ArowM = Mm + (size_t)(i0 + m) * Vc + 2 * hh;

  #pragma unroll 1
  for (int it = 0; it < SINK_ITERS; ++it) {
    // ---- phase A: u = a / (K @ v + 1e-16) ----
    {
      v8f c = {0.f,0.f,0.f,0.f,0.f,0.f,0.f,0.f};
      #pragma unroll 1
      for (int jc = 0; jc < Vc; jc += 32) {
        __syncthreads();
        for (int idx = threadIdx.x; idx < 32 * NP; idx += 256)
          Vsh[idx] = vD[(size_t)(jc + (idx >> 6)) * NP + (idx & 63)];
        __syncthreads();
        if (valid) {
          __builtin_prefetch(Arow + jc + 32, 0, 3);   // stream next Km chunk
          for (int js = 0; js < 32; js += 4) {
            v2f a = loadv2(Arow + jc + js);
            v2f b; b.x = Vsh[(js + 2 * hh) * NP + n0 + m];
                   b.y = Vsh[(js + 2 * hh + 1) * NP + n0 + m];
            c = wmma4(a, b, c);
          }
        }
      }
      if (valid) {
        for (int e = 0; e < 8; ++e) {
          size_t o = (size_t)(i0 + e + hh * 8) * NP + n0 + m;
          uD[o] = aD[o] / (c[e] + 1e-16f);
        }
      }
    }
    grid_sync(cnt, gen, SINK_BLOCKS);
    // ---- phase B: v = b / (K^T @ u + 1e-16) ----
    {
      v8f c = {0.f,0.f,0.f,0.f,0.f,0.f,0.f,0.f};
      #pragma unroll 1
      for (int jc = 0; jc < Vc; jc += 32) {
        __syncthreads();
        for (int idx = threadIdx.x; idx < 32 * NP; idx += 256)
          Vsh[idx] = uD[(size_t)(jc + (idx >> 6)) * NP + (idx & 63)];
        __syncthreads();
        if (valid) {
          for (int js = 0; js < 32; js += 4) {
            int ka = jc + js + 2 * hh;
            v2f a; a.x = Km[(size_t)ka * Vc + i0 + m];
                   a.y = Km[(size_t)(ka + 1) * Vc + i0 + m];
            v2f b; b.x = Vsh[(js + 2 * hh) * NP + n0 + m];
                   b.y = Vsh[(js + 2 * hh + 1) * NP + n0 + m];
            c = wmma4(a, b, c);
          }
        }
      }
      if (valid) {
        for (int e = 0; e < 8; ++e) {
          size_t o = (size_t)(i0 + e + hh * 8) * NP + n0 + m;
          vD[o] = bD[o] / (c[e] + 1e-16f);
        }
      }
    }
    grid_sync(cnt, gen, SINK_BLOCKS);
  }

  // ---- final: u = a/(K v + eps); s_loss = sum(u * ((K*M) @ v)) ----
  v8f ckv = {0.f,0.f,0.f,0.f,0.f,0.f,0.f,0.f};
  v8f ckm = {0.f,0.f,0.f,0.f,0.f,0.f,0.f,0.f};
  #pragma unroll 1
  for (int jc = 0; jc < Vc; jc += 32) {
    __syncthreads();
    for (int idx = threadIdx.x; idx < 32 * NP; idx += 256)
      Vsh[idx] = vD[(size_t)(jc + (idx >> 6)) * NP + (idx & 63)];
    __syncthreads();
    if (valid) {
      __builtin_prefetch(Arow + jc + 32, 0, 3);
      __builtin_prefetch(ArowM + jc + 32, 0, 3);
      for (int js = 0; js < 32; js += 4) {
        v2f a  = loadv2(Arow  + jc + js);
        v2f mm = loadv2(ArowM + jc + js);
        v2f am; am.x = a.x * mm.x;  am.y = a.y * mm.y;
        v2f b;  b.x = Vsh[(js + 2 * hh) * NP + n0 + m];
                b.y = Vsh[(js + 2 * hh + 1) * NP + n0 + m];
        ckv = wmma4(a,  b, ckv);
        ckm = wmma4(am, b, ckm);
      }
    }
  }
  float local = 0.f;
  if (valid) {
    for (int e = 0; e < 8; ++e) {
      size_t o = (size_t)(i0 + e + hh * 8) * NP + n0 + m;
      float u = aD[o] / (ckv[e] + 1e-16f);
      local += u * ckm[e];
    }
  }
  red[threadIdx.x] = local; __syncthreads();
  for (int s = 128; s > 0; s >>= 1) { if (threadIdx.x < s) red[threadIdx.x] += red[threadIdx.x + s]; __syncthreads(); }
  if (threadIdx.x == 0) part[blockIdx.x] = red[0];
}

// ---------------- final scalar combine ----------------
__global__ void k_final(const float* __restrict__ la, const float* __restrict__ lb,
                        const float* __restrict__ part, float* __restrict__ out) {
  __shared__ float red[256];
  int t = threadIdx.x;
  float s = 0.f;
  for (int i = t; i < Bc; i += 256) s += la[i] + lb[i];
  float elbo_sum = blockReduceSum256(s, red);
  float sp = (t < SINK_BLOCKS) ? part[t] : 0.f;
  float sloss = blockReduceSum256(sp, red);
  if (t == 0) out[0] = elbo_sum / (float)Bc + 0.1f * sloss;
}

// ---------------- host side ----------------
static inline void launch_gemm(const float* A, const float* Bm, const float* bias,
                               float* C, float* C2, int Mr, int Kd, int Nc, int mode,
                               hipStream_t s) {
  int tiles = ((Mr + 15) / 16) * ((Nc + 15) / 16);
  int blocks = (tiles + 7) / 8;
  gemm_wmma<<<blocks, 256, 0, s>>>(A, Bm, bias, C, C2, Mr, Kd, Nc, mode);
}

extern "C" void kernel_launch(void* const* d_in, const int* in_sizes, int n_in,
                              void* d_out, int out_size, void* d_ws, size_t ws_size,
                              hipStream_t stream) {
  (void)in_sizes; (void)n_in; (void)out_size; (void)ws_size;
  const float* IN[30];
  for (int i = 0; i < 30; ++i) IN[i] = (const float*)d_in[i];
  float* ws  = (float*)d_ws;
  float* out = (float*)d_out;

  float* h1   = ws + OF_H1;  float* h2  = ws + OF_H2;
  float* mu   = ws + OF_MU;  float* lv  = ws + OF_LV;
  float* th   = ws + OF_TH;  float* y   = ws + OF_Y;
  float* my   = ws + OF_MY;  float* vy  = ws + OF_VY;
  float* mmu  = ws + OF_MMU; float* vmu = ws + OF_VMU;
  float* mlv  = ws + OF_MLV; float* vlv = ws + OF_VLV;
  float* kld  = ws + OF_KLD;
  float* la   = ws + OF_LA;  float* lb  = ws + OF_LB;
  float* part = ws + OF_PART;
  int*   sync = (int*)(ws + OF_SYNC);
  float* Mb   = ws + OF_M;   float* Kmb = ws + OF_KM;
  float* na   = ws + OF_NA;  float* nb  = ws + OF_NB;
  float* aD   = ws + OF_AD;  float* bD  = ws + OF_BD;
  float* uD   = ws + OF_UD;  float* vD  = ws + OF_VD;

  // ---- encoder/decoder for both languages ----
  for (int enc = 0; enc < 2; ++enc) {
    const float* x   = IN[0 + enc];
    const float* eps = IN[2 + enc];
    int base = (enc == 0) ? 4 : 14;
    const float* W1  = IN[base + 0]; const float* b1  = IN[base + 1];
    const float* W2  = IN[base + 2]; const float* b2  = IN[base + 3];
    const float* Wmu = IN[base + 4]; const float* bmu = IN[base + 5];
    const float* Wlv = IN[base + 6]; const float* blv = IN[base + 7];
    const float* mbn = IN[base + 8]; const float* lbn = IN[base + 9];
    const float* dbn = IN[24 + enc];
    const float* phi = IN[26 + enc];
    float* rowloss   = (enc == 0) ? la : lb;

    launch_gemm(x,  W1,  b1,  h1, nullptr, Bc, Vc, Hc, 2, stream);   // softplus
    launch_gemm(h1, W2,  b2,  h2, nullptr, Bc, Hc, Hc, 2, stream);   // softplus
    launch_gemm(h2, Wmu, bmu, mu, nullptr, Bc, Hc, Kc, 1, stream);   // +bias
    launch_gemm(h2, Wlv, blv, lv, nullptr, Bc, Hc, Kc, 1, stream);   // +bias
    k_colstats<<<1, 256, 0, stream>>>(mu, Bc, Kc, mmu, vmu);
    k_colstats<<<1, 256, 0, stream>>>(lv, Bc, Kc, mlv, vlv);
    k_theta<<<Bc, 64, 0, stream>>>(mu, lv, mmu, vmu, mlv, vlv, mbn, lbn, eps, th, kld);
    launch_gemm(th, phi, nullptr, y, nullptr, Bc, Kc, Vc, 0, stream); // plain
    k_colstats<<<(Vc + 255) / 256, 256, 0, stream>>>(y, Bc, Vc, my, vy);
    k_rec<<<Bc, 256, 0, stream>>>(y, x, my, vy, dbn, kld, rowloss);
  }

  // ---- cosine cost matrix + Kmat ----
  k_rownorm<<<Vc, 128, 0, stream>>>(IN[28], na);
  k_rownorm<<<Vc, 128, 0, stream>>>(IN[29], nb);
  launch_gemm(na, nb, nullptr, Mb, Kmb, Vc, Ec, Vc, 3, stream);       // cosine mode

  // ---- sinkhorn ----
  k_initsink<<<(4 * Vc * NP + 255) / 256, 256, 0, stream>>>(aD, part, sync);
  k_colsoftmax<<<Kc, 256, 0, stream>>>(IN[26], aD);
  k_colsoftmax<<<Kc, 256, 0, stream>>>(IN[27], bD);
  k_sinkhorn<<<SINK_BLOCKS, 256, 0, stream>>>(Kmb, Mb, aD, bD, uD, vD, part, sync);

  // ---- combine ----
  k_final<<<1, 256, 0, stream>>>(la, lb, part, out);
}